// GraphConvolution_10007273800446
// MI455X (gfx1250) — compile-verified
//
#include <hip/hip_runtime.h>
#include <hip/hip_bf16.h>

// ---------------------------------------------------------------------------
// GraphConvolution on MI455X (gfx1250, wave32, WMMA).
//
//   w          = edge_features @ p^T                      [E]
//   multiplier = (T * w[None,:]) @ T^T                    [N,N]   (550 GFLOP)
//   adjA       = (I + (1-I)*multiplier) * adj_v           [N,N]
//   G          = H_v @ weight                             [N,dout]
//   out        = adjA @ G + bias                          [N,dout]
//
// Memory-bound (T = 256MB, 23.3 TB/s HBM, 192MB L2) -> full fp32 precision
// via V_WMMA_F32_16X16X4_F32.  Both LDS tiles are K-major so every WMMA
// fragment is one aligned ds_load_b64 (no register shuffles), and the hot
// B-tile copy of kernel 3 uses GLOBAL_LOAD_ASYNC_TO_LDS_B128 when available.
// ---------------------------------------------------------------------------

typedef float v2f __attribute__((ext_vector_type(2)));
typedef float v8f __attribute__((ext_vector_type(8)));

#define NN    4096
#define EE    16384
#define DIN   1024
#define DOUT  1024

#define BM 64
#define BN 64
#define BK 64
#define KPAD 4          // row stride 68 floats: bank = (4*row + k) % 64, conflict free
#define LDK (BK + KPAD)
#define THREADS 128     // 4 wave32s, 2x2 wave grid, 32x32 per wave

#if __has_builtin(__builtin_amdgcn_global_load_async_to_lds_b128)
#define HAVE_ASYNC_LDS 1
// Builtin signature (from clang diagnostic): param 0 is AS1 (global) pointer
// to a 16-byte int vector, param 1 is the LDS-side pointer, then imm offset
// and imm cpol.
typedef __attribute__((__vector_size__(4 * sizeof(int)))) int v4i_t;
typedef __attribute__((address_space(1))) v4i_t GV4i;
typedef __attribute__((address_space(3))) v4i_t LV4i;

__device__ __forceinline__ void async_copy_b128(const float* gsrc, float* ldst) {
    __builtin_amdgcn_global_load_async_to_lds_b128((GV4i*)gsrc, (LV4i*)ldst, 0, 0);
}
__device__ __forceinline__ void wait_async_zero() {
#if __has_builtin(__builtin_amdgcn_s_wait_asynccnt)
    __builtin_amdgcn_s_wait_asynccnt(0);
#else
    asm volatile("s_wait_asynccnt 0x0" ::: "memory");
#endif
}
#else
#define HAVE_ASYNC_LDS 0
#endif

__device__ __forceinline__ v8f wmma_f32(v2f a, v2f b, v8f c) {
    // (neg_a, A, neg_b, B, c_mod, C, reuse_a, reuse_b)
    return __builtin_amdgcn_wmma_f32_16x16x4_f32(false, a, false, b,
                                                 (short)0, c, false, false);
}

// ---------------------------------------------------------------------------
// w[e] = edge_features[e,:] . p   (de = 3)
// ---------------------------------------------------------------------------
__global__ void edge_w_kernel(const float* __restrict__ ef,
                              const float* __restrict__ p,
                              float* __restrict__ w, int E) {
    int e = blockIdx.x * blockDim.x + threadIdx.x;
    if (e < E) {
        w[e] = ef[3 * e + 0] * p[0] + ef[3 * e + 1] * p[1] + ef[3 * e + 2] * p[2];
    }
}

// ---------------------------------------------------------------------------
// Generic tiled WMMA fp32 GEMM:  C = A(MxK) * B(KxN)  [+ epilogue]
//   SCALE_A : A columns scaled by wvec[k]       (for T * w[None,:])
//   B_TRANS : B[k][n] = Bmem[n*ldb + k]         (for @ T^T)
//   EPI = 0 : C = acc
//   EPI = 1 : C[i][j] = (i==j ? 1 : acc) * adj[i][j]
//   EPI = 2 : C[i][j] = acc + bias[j]
// LDS layout: As[m][k], Bs[n][k] (both K-major, padded) so each 16x16x4
// fragment is a single aligned 8-byte LDS read.
// ---------------------------------------------------------------------------
template <int SCALE_A, int B_TRANS, int EPI>
__global__ __launch_bounds__(THREADS)
void gemm_wmma_f32(const float* __restrict__ A, int lda,
                   const float* __restrict__ B, int ldb,
                   float* __restrict__ C, int ldc,
                   int K,
                   const float* __restrict__ wvec,
                   const float* __restrict__ adj, int ldadj,
                   const float* __restrict__ bias) {
    __shared__ __align__(16) float As[BM * LDK];
    __shared__ __align__(16) float Bs[BN * LDK];

    const int m0   = blockIdx.y * BM;
    const int n0   = blockIdx.x * BN;
    const int tid  = threadIdx.x;
    const int lane = tid & 31;
    const int wave = tid >> 5;
    const int wm   = (wave >> 1) * 32;   // wave M offset in block tile
    const int wn   = (wave & 1) * 32;    // wave N offset in block tile
    const int lr   = lane & 15;          // M index (A frag) / N index (B frag)
    const int hi   = lane >> 4;          // lane-half selects K pair {0,1} vs {2,3}

    v8f acc00 = {}, acc01 = {}, acc10 = {}, acc11 = {};

    for (int k0 = 0; k0 < K; k0 += BK) {
        // ---- stage A tile: As[r][c] = A[m0+r][k0+c] (* wvec[k0+c]) ----
#pragma unroll
        for (int i = 0; i < 8; ++i) {
            int idx = tid + i * THREADS;          // 0..1023 float4 slots
            int r   = idx >> 4;
            int c4  = (idx & 15) << 2;
            float4 va = *(const float4*)(A + (long)(m0 + r) * lda + k0 + c4);
            if (SCALE_A) {
                float4 sw = *(const float4*)(wvec + k0 + c4);
                va.x *= sw.x; va.y *= sw.y; va.z *= sw.z; va.w *= sw.w;
            }
            *(float4*)(&As[r * LDK + c4]) = va;
        }
        // ---- stage B tile: Bs[n][k] ----
        if (B_TRANS) {
            // Bs[n][k] = Bmem[(n0+n)*ldb + k0+k]  -- straight row copy of T
#pragma unroll
            for (int i = 0; i < 8; ++i) {
                int idx = tid + i * THREADS;
                int nn  = idx >> 4;
                int k4  = (idx & 15) << 2;
#if HAVE_ASYNC_LDS
                async_copy_b128(B + (long)(n0 + nn) * ldb + k0 + k4,
                                &Bs[nn * LDK + k4]);
#else
                float4 vb = *(const float4*)(B + (long)(n0 + nn) * ldb + k0 + k4);
                *(float4*)(&Bs[nn * LDK + k4]) = vb;
#endif
            }
#if HAVE_ASYNC_LDS
            wait_async_zero();
#endif
        } else {
            // Bs[n][k] = Bmem[(k0+k)*ldb + n0+n]  -- transpose into LDS
#pragma unroll
            for (int i = 0; i < 8; ++i) {
                int idx = tid + i * THREADS;
                int kk  = idx >> 4;
                int c4  = (idx & 15) << 2;
                float4 vb = *(const float4*)(B + (long)(k0 + kk) * ldb + n0 + c4);
                Bs[(c4 + 0) * LDK + kk] = vb.x;
                Bs[(c4 + 1) * LDK + kk] = vb.y;
                Bs[(c4 + 2) * LDK + kk] = vb.z;
                Bs[(c4 + 3) * LDK + kk] = vb.w;
            }
        }
        __syncthreads();

        // ---- CDNA5 speculative prefetch of the next k-panel ----
        if (k0 + BK < K) {
            const float* pa = A + (long)(m0 + (tid >> 1)) * lda + (k0 + BK) + ((tid & 1) << 5);
            __builtin_prefetch(pa, 0, 1);
            const float* pb = B_TRANS
                ? B + (long)(n0 + (tid >> 1)) * ldb + (k0 + BK) + ((tid & 1) << 5)
                : B + (long)(k0 + BK + (tid >> 1)) * ldb + n0 + ((tid & 1) << 5);
            __builtin_prefetch(pb, 0, 1);
        }

        // ---- 16 x (2x2) WMMA f32 16x16x4; every fragment = 1 ds_load_b64 ----
#pragma unroll
        for (int kk = 0; kk < BK; kk += 4) {
            const int ka = kk + 2 * hi;   // K={0,1} lanes 0-15, K={2,3} lanes 16-31
            v2f a0 = *(const v2f*)(&As[(wm + lr)      * LDK + ka]);
            v2f a1 = *(const v2f*)(&As[(wm + 16 + lr) * LDK + ka]);
            v2f b0 = *(const v2f*)(&Bs[(wn + lr)      * LDK + ka]);
            v2f b1 = *(const v2f*)(&Bs[(wn + 16 + lr) * LDK + ka]);
            acc00 = wmma_f32(a0, b0, acc00);
            acc01 = wmma_f32(a0, b1, acc01);
            acc10 = wmma_f32(a1, b0, acc10);
            acc11 = wmma_f32(a1, b1, acc11);
        }
        __syncthreads();
    }

    // ---- epilogue; C/D layout: VGPR v -> M = v + 8*hi, N = lane&15 ----
    v8f accs[2][2] = {{acc00, acc01}, {acc10, acc11}};
#pragma unroll
    for (int ti = 0; ti < 2; ++ti) {
#pragma unroll
        for (int tj = 0; tj < 2; ++tj) {
            const int rbase = m0 + wm + ti * 16 + hi * 8;
            const int col   = n0 + wn + tj * 16 + lr;
#pragma unroll
            for (int v = 0; v < 8; ++v) {
                const int row = rbase + v;
                float val = accs[ti][tj][v];
                if (EPI == 1) {
                    float a = adj[(long)row * ldadj + col];
                    val = (row == col ? 1.0f : val) * a;
                } else if (EPI == 2) {
                    val += bias[col];
                }
                C[(long)row * ldc + col] = val;
            }
        }
    }
}

// ---------------------------------------------------------------------------
extern "C" void kernel_launch(void* const* d_in, const int* in_sizes, int n_in,
                              void* d_out, int out_size, void* d_ws, size_t ws_size,
                              hipStream_t stream) {
    const float* H    = (const float*)d_in[0];   // [N, din]
    const float* ef   = (const float*)d_in[1];   // [E, 3]
    const float* adj  = (const float*)d_in[2];   // [N, N]
    const float* T    = (const float*)d_in[3];   // [N, E]
    const float* Wt   = (const float*)d_in[4];   // [din, dout]
    const float* bias = (const float*)d_in[5];   // [dout]
    const float* p    = (const float*)d_in[6];   // [3]
    float* out = (float*)d_out;                  // [N, dout]

    char* ws = (char*)d_ws;
    float* wvec = (float*)ws;                                    // E floats (64 KB)
    float* G    = (float*)(ws + (size_t)EE * sizeof(float));     // N*dout  (16 MB)
    float* adjA = (float*)(ws + (size_t)EE * sizeof(float)
                              + (size_t)NN * DOUT * sizeof(float)); // N*N (64 MB)

    dim3 blk(THREADS);

    // 1) per-edge scalar weights
    edge_w_kernel<<<EE / 256, 256, 0, stream>>>(ef, p, wvec, EE);

    // 2) G = H_v @ weight                  (M=4096, K=1024, N=1024)
    gemm_wmma_f32<0, 0, 0><<<dim3(DOUT / BN, NN / BM), blk, 0, stream>>>(
        H, DIN, Wt, DOUT, G, DOUT, DIN, nullptr, nullptr, 0, nullptr);

    // 3) adjA = ((T*w) @ T^T, diag->1) * adj_v   (M=N=4096, K=16384)
    gemm_wmma_f32<1, 1, 1><<<dim3(NN / BN, NN / BM), blk, 0, stream>>>(
        T, EE, T, EE, adjA, NN, EE, wvec, adj, NN, nullptr);

    // 4) out = adjA @ G + bias             (M=4096, K=4096, N=1024)
    gemm_wmma_f32<0, 0, 2><<<dim3(DOUT / BN, NN / BM), blk, 0, stream>>>(
        adjA, NN, G, DOUT, out, DOUT, NN, nullptr, nullptr, 0, bias);
}